// Topo_60653528154131
// MI455X (gfx1250) — compile-verified
//
#include <hip/hip_runtime.h>
#include <math.h>

typedef __attribute__((ext_vector_type(2))) float v2f;
typedef __attribute__((ext_vector_type(8))) float v8f;

#define N_PTS 256   // batch N -> points in the distance matrix
#define N_C   3
#define N_T   256
#define N_V   25
#define CT    768   // C*T
#define KP    28    // K=25 padded to multiple of 4 for V_WMMA_F32_16X16X4_F32
#define N_EL  64

// workspace layout (float offsets)
#define WS_M    0                    // m   [256*25]
#define WS_FNP  (WS_M + N_PTS*N_V)   // fnp [256*28] zero-padded
#define WS_SQ   (WS_FNP + N_PTS*KP)  // sq  [256]
#define WS_D    (WS_SQ + N_PTS)      // D   [256*256]

// ---------------------------------------------------------------------------
// Kernel 1: m[n,v] = mean over (c,t) of x[n,c,t,v].  One block per n.
// Deterministic: shuffle reduce within wave32, fixed-order LDS partial sum.
// ---------------------------------------------------------------------------
__global__ void mean_ct_kernel(const float* __restrict__ x, float* __restrict__ m)
{
    const int n = blockIdx.x;
    const int t = threadIdx.x;                 // 256 threads
    const float* xb = x + (size_t)n * (CT * N_V);

    float acc[N_V];
#pragma unroll
    for (int v = 0; v < N_V; ++v) acc[v] = 0.f;

#pragma unroll
    for (int rr = 0; rr < 3; ++rr) {           // 768 rows / 256 threads
        const float* row = xb + (t + rr * 256) * N_V;
#pragma unroll
        for (int v = 0; v < N_V; ++v) acc[v] += row[v];
    }

    // wave32 reduction
#pragma unroll
    for (int v = 0; v < N_V; ++v) {
        for (int off = 16; off > 0; off >>= 1)
            acc[v] += __shfl_down(acc[v], off, 32);
    }

    __shared__ float part[8][N_V];
    const int wave = t >> 5, lane = t & 31;
    if (lane == 0) {
#pragma unroll
        for (int v = 0; v < N_V; ++v) part[wave][v] = acc[v];
    }
    __syncthreads();
    if (t < N_V) {
        float s = 0.f;
#pragma unroll
        for (int w = 0; w < 8; ++w) s += part[w][t];
        m[n * N_V + t] = s * (1.0f / (float)CT);
    }
}

// ---------------------------------------------------------------------------
// Kernel 2: feat[n,j] = ||m[n,:] - m[n,j]||, global min-max normalize,
// write zero-padded fn[256][28] and row squared-norms sq[256].  1 block.
// ---------------------------------------------------------------------------
__global__ void feat_kernel(const float* __restrict__ m,
                            float* __restrict__ fnp, float* __restrict__ sq)
{
    const int n = threadIdx.x;                 // 256 threads, thread == point
    float ml[N_V];
#pragma unroll
    for (int v = 0; v < N_V; ++v) ml[v] = m[n * N_V + v];

    float feat[N_V];
    float lmin = __builtin_inff(), lmax = -__builtin_inff();
#pragma unroll
    for (int j = 0; j < N_V; ++j) {
        float s = 0.f;
#pragma unroll
        for (int i = 0; i < N_V; ++i) { float d = ml[i] - ml[j]; s += d * d; }
        float f = sqrtf(s);
        feat[j] = f;
        lmin = fminf(lmin, f);
        lmax = fmaxf(lmax, f);
    }

    __shared__ float smin[N_PTS], smax[N_PTS];
    smin[n] = lmin; smax[n] = lmax;
    __syncthreads();
    for (int s = 128; s > 0; s >>= 1) {
        if (n < s) {
            smin[n] = fminf(smin[n], smin[n + s]);
            smax[n] = fmaxf(smax[n], smax[n + s]);
        }
        __syncthreads();
    }
    const float mn = smin[0];
    const float inv = 1.0f / (smax[0] - mn);

    float sl = 0.f;
#pragma unroll
    for (int j = 0; j < N_V; ++j) {
        float f = (feat[j] - mn) * inv;
        fnp[n * KP + j] = f;
        sl += f * f;
    }
#pragma unroll
    for (int j = N_V; j < KP; ++j) fnp[n * KP + j] = 0.f;
    sq[n] = sl;
}

// ---------------------------------------------------------------------------
// Kernel 3: D = sqrt(max(sq_a + sq_b - 2*G, 0)) with G = fn @ fn^T computed
// by V_WMMA_F32_16X16X4_F32.  One wave (32 lanes, EXEC all-1s) per 16x16 tile.
// A 16x4 f32 layout: lane L -> row L&15, K-offset (L>>4)*2 (2 VGPRs).
// B 4x16 layout mirrors it; since B = fn^T the lane loads are symmetric.
// C/D layout: VGPR r -> M = r + 8*(L>=16), N = L&15.
// ---------------------------------------------------------------------------
__global__ void dist_wmma_kernel(const float* __restrict__ fnp,
                                 const float* __restrict__ sq,
                                 float* __restrict__ D)
{
    const int tile = blockIdx.x;               // 256 tiles (16x16 grid)
    const int bi = tile >> 4, bj = tile & 15;
    const int L  = threadIdx.x;                // 32 lanes
    const int mrow = L & 15;
    const int koff = (L >> 4) << 1;

    const float* arow = fnp + (bi * 16 + mrow) * KP + koff;
    const float* brow = fnp + (bj * 16 + mrow) * KP + koff;

    v8f c = {0.f, 0.f, 0.f, 0.f, 0.f, 0.f, 0.f, 0.f};
#pragma unroll
    for (int k0 = 0; k0 < KP; k0 += 4) {
        v2f a, b;
        a.x = arow[k0]; a.y = arow[k0 + 1];
        b.x = brow[k0]; b.y = brow[k0 + 1];
        // 8 args: (neg_a, A, neg_b, B, c_mod, C, reuse_a, reuse_b)
        c = __builtin_amdgcn_wmma_f32_16x16x4_f32(
                false, a, false, b, (short)0, c, false, false);
    }

    const int mhalf = (L >> 4) * 8;
    const int col = bj * 16 + (L & 15);
    const float sqc = sq[col];
#pragma unroll
    for (int r = 0; r < 8; ++r) {
        const int row = bi * 16 + r + mhalf;
        const float g  = c[r];
        const float d2 = sq[row] + sqc - 2.f * g;
        D[row * N_PTS + col] = (d2 > 0.f) ? sqrtf(d2) : 0.f;
    }
}

// ---------------------------------------------------------------------------
// Kernel 4: Prim's MST (255 sequential steps, LDS argmin tree with
// lowest-index tie-break matching jnp.argmin) + structure-element layer.
// Single block of 256 threads.
// ---------------------------------------------------------------------------
__global__ void mst_struct_kernel(const float* __restrict__ D,
                                  const float* __restrict__ centres,
                                  const float* __restrict__ sharp,
                                  float* __restrict__ out)
{
    __shared__ float dist[N_PTS];
    __shared__ int   intree[N_PTS];
    __shared__ float svals[N_PTS];
    __shared__ int   sidx[N_PTS];
    __shared__ float deaths[N_PTS - 1];

    const int t = threadIdx.x;
    dist[t]   = D[t];                          // row 0
    intree[t] = (t == 0) ? 1 : 0;
    __syncthreads();

    for (int step = 0; step < N_PTS - 1; ++step) {
        svals[t] = intree[t] ? __builtin_inff() : dist[t];
        sidx[t]  = t;
        __syncthreads();
        for (int s = 128; s > 0; s >>= 1) {
            if (t < s) {
                const float vo = svals[t + s]; const int io = sidx[t + s];
                if (vo < svals[t] || (vo == svals[t] && io < sidx[t])) {
                    svals[t] = vo; sidx[t] = io;
                }
            }
            __syncthreads();
        }
        const int   j = sidx[0];
        const float w = svals[0];
        if (t == 0) deaths[step] = w;
        dist[t] = fminf(dist[t], D[j * N_PTS + t]);
        if (t == j) intree[t] = 1;
        __syncthreads();
    }

    // out[k] = sum_i exp(-((0-c_k0)^2 s_k0^2 + (d_i-c_k1)^2 s_k1^2))
    if (t < N_EL) {
        const float c0 = centres[t * 2 + 0], c1 = centres[t * 2 + 1];
        const float s0 = sharp[t * 2 + 0],   s1 = sharp[t * 2 + 1];
        const float term0 = c0 * c0 * s0 * s0;
        const float s1sq  = s1 * s1;
        float acc = 0.f;
        for (int i = 0; i < N_PTS - 1; ++i) {
            const float dd = deaths[i] - c1;
            acc += expf(-(term0 + dd * dd * s1sq));
        }
        out[t] = acc;
    }
}

// ---------------------------------------------------------------------------
extern "C" void kernel_launch(void* const* d_in, const int* in_sizes, int n_in,
                              void* d_out, int out_size, void* d_ws, size_t ws_size,
                              hipStream_t stream)
{
    const float* x       = (const float*)d_in[0];   // [256,3,256,25]
    const float* centres = (const float*)d_in[1];   // [64,2]
    const float* sharp   = (const float*)d_in[2];   // [64,2]
    float* out = (float*)d_out;                     // [64]
    float* ws  = (float*)d_ws;

    float* m   = ws + WS_M;
    float* fnp = ws + WS_FNP;
    float* sq  = ws + WS_SQ;
    float* D   = ws + WS_D;

    mean_ct_kernel<<<N_PTS, 256, 0, stream>>>(x, m);
    feat_kernel<<<1, 256, 0, stream>>>(m, fnp, sq);
    dist_wmma_kernel<<<N_PTS, 32, 0, stream>>>(fnp, sq, D);
    mst_struct_kernel<<<1, 256, 0, stream>>>(D, centres, sharp, out);
}